// GATLayer_30648886624350
// MI455X (gfx1250) — compile-verified
//
#include <hip/hip_runtime.h>
#include <math.h>
#include <stdint.h>

#define N_NODES 8192
#define F_IN    128
#define F_OUT   64
#define ALPHA   0.2f
#define NEG_INF -9e15f

typedef __attribute__((ext_vector_type(2))) float v2f;
typedef __attribute__((ext_vector_type(8))) float v8f;
typedef __attribute__((ext_vector_type(4))) unsigned int v4u;
typedef __attribute__((ext_vector_type(8))) int v8i;
typedef __attribute__((ext_vector_type(4))) int v4i;

__device__ __forceinline__ float lrelu(float x) { return fmaxf(x, ALPHA * x); }

// ---------------------------------------------------------------------------
// Kernel 1: Wh = h @ W   [8192x64],  src = Wh @ a[:64],  dst = Wh @ a[64:]
// ---------------------------------------------------------------------------
__global__ void __launch_bounds__(256) gat_proj(const float* __restrict__ h,
                                                const float* __restrict__ W,
                                                const float* __restrict__ a,
                                                float* __restrict__ Wh,
                                                float* __restrict__ src,
                                                float* __restrict__ dst) {
    __shared__ float sh_h[4 * F_IN];
    __shared__ float sh_as[F_OUT];
    __shared__ float sh_ad[F_OUT];
    __shared__ float red[256];

    int t = threadIdx.x;
    int rowBase = blockIdx.x * 4;

    for (int i = t; i < 4 * F_IN; i += 256) sh_h[i] = h[(size_t)rowBase * F_IN + i];
    if (t < F_OUT) { sh_as[t] = a[t]; sh_ad[t] = a[F_OUT + t]; }
    __syncthreads();

    int r = t >> 6;   // local row 0..3
    int f = t & 63;   // feature 0..63
    float acc = 0.f;
    #pragma unroll 8
    for (int k = 0; k < F_IN; ++k)
        acc = fmaf(sh_h[r * F_IN + k], W[k * F_OUT + f], acc);

    int row = rowBase + r;
    Wh[(size_t)row * F_OUT + f] = acc;

    red[t] = acc * sh_as[f];
    __syncthreads();
    for (int s = 32; s >= 1; s >>= 1) {
        if (f < s) red[t] += red[t + s];
        __syncthreads();
    }
    if (f == 0) src[row] = red[t];
    __syncthreads();

    red[t] = acc * sh_ad[f];
    __syncthreads();
    for (int s = 32; s >= 1; s >>= 1) {
        if (f < s) red[t] += red[t + s];
        __syncthreads();
    }
    if (f == 0) dst[row] = red[t];
}

// ---------------------------------------------------------------------------
// Kernel 2: per-row softmax stats: rowM = max_j e_ij, rowS = sum_j exp(e - M)
// ---------------------------------------------------------------------------
__global__ void __launch_bounds__(256) gat_rowstats(const int* __restrict__ adj,
                                                    const float* __restrict__ src,
                                                    const float* __restrict__ dst,
                                                    float* __restrict__ rowM,
                                                    float* __restrict__ rowS) {
    __shared__ float red[256];
    int t = threadIdx.x;
    int row = blockIdx.x;
    const int* arow = adj + (size_t)row * N_NODES;
    float sv = src[row];

    float mloc = -3.4e38f;
    for (int j = t; j < N_NODES; j += 256) {
        float e = lrelu(sv + dst[j]);
        e = (arow[j] > 0) ? e : NEG_INF;
        mloc = fmaxf(mloc, e);
    }
    red[t] = mloc;
    __syncthreads();
    for (int s = 128; s >= 1; s >>= 1) {
        if (t < s) red[t] = fmaxf(red[t], red[t + s]);
        __syncthreads();
    }
    float M = red[0];
    __syncthreads();

    float ssum = 0.f;
    for (int j = t; j < N_NODES; j += 256) {
        float e = lrelu(sv + dst[j]);
        e = (arow[j] > 0) ? e : NEG_INF;
        ssum += __expf(e - M);
    }
    red[t] = ssum;
    __syncthreads();
    for (int s = 128; s >= 1; s >>= 1) {
        if (t < s) red[t] += red[t + s];
        __syncthreads();
    }
    if (t == 0) { rowM[row] = M; rowS[row] = red[0]; }
}

// ---------------------------------------------------------------------------
// Kernel 3: out = ELU( softmax(e) @ Wh ) via V_WMMA_F32_16X16X4_F32 with
// TDM (TENSOR_LOAD_TO_LDS) double-buffered Wh staging.
// D# loads a 64x64 f32 tile, padding LDS by 16 dwords every 64 dwords so the
// LDS row stride is 80 dwords (conflict-free ds_load phases).
// ---------------------------------------------------------------------------
#define CHUNK      64
#define K3_THREADS 64
#define WH_STRIDE  80        // 64 data + 16 pad dwords per row (TDM pad)
#define NCHUNKS    (N_NODES / CHUNK)

__device__ __forceinline__ void tdm_load_tile(const float* gtile, unsigned int lds_off) {
    unsigned long long ga = (unsigned long long)(uintptr_t)gtile;
    v4u g0;
    g0[0] = 1u;                                        // count=1, user descriptor
    g0[1] = lds_off;                                   // lds_addr (bytes)
    g0[2] = (unsigned int)(ga & 0xffffffffu);          // global_addr[31:0]
    g0[3] = (unsigned int)((ga >> 32) & 0x1ffffffu)    // global_addr[56:32]
          | (2u << 30);                                // type=2 (image)
    v8i g1;
    g1[0] = (int)((2u << 16)      // data_size = 4B
                | (1u << 20)      // pad_enable
                | (5u << 22)      // pad_interval: 64 dwords
                | (15u << 25));   // pad_amount: 16 dwords
    g1[1] = (int)(F_OUT << 16);            // tensor_dim0[15:0]=64
    g1[2] = (int)(N_NODES << 16);          // dim0 hi=0 | tensor_dim1[15:0]=8192
    g1[3] = (int)(F_OUT << 16);            // dim1 hi=0 | tile_dim0=64
    g1[4] = (int)CHUNK;                    // tile_dim1=64, tile_dim2=0
    g1[5] = (int)F_OUT;                    // tensor_dim0_stride=64
    g1[6] = 0;                             // stride0 hi | tensor_dim1_stride lo (unused, 2D)
    g1[7] = 0;
    v4i z4 = {0, 0, 0, 0};
    v8i z8 = {0, 0, 0, 0, 0, 0, 0, 0};
    __builtin_amdgcn_tensor_load_to_lds(g0, g1, z4, z4, z8, 0);   // 6-arg form (clang-23)
}

__global__ void __launch_bounds__(K3_THREADS) gat_aggregate(const int* __restrict__ adj,
                                                            const float* __restrict__ Wh,
                                                            const float* __restrict__ src,
                                                            const float* __restrict__ dst,
                                                            const float* __restrict__ rowM,
                                                            const float* __restrict__ rowS,
                                                            float* __restrict__ out) {
    __shared__ __align__(16) float sh_wh[2][CHUNK * WH_STRIDE];   // 2 x 20 KB

    int t = threadIdx.x;
    int wave = t >> 5;
    int lane = t & 31;
    int half = lane >> 4;      // 0: K={0,1}, 1: K={2,3}
    int n = lane & 15;         // N column within 16-wide tile / A row index

    int rowBase = blockIdx.x * 32 + wave * 16;
    int mrow = rowBase + n;
    float src_m = src[mrow];
    float max_m = rowM[mrow];
    const int* arow = adj + (size_t)mrow * N_NODES;

    v8f c0 = {}, c1 = {}, c2 = {}, c3 = {};

    // Prologue: wave 0 pulls chunk 0 via the Tensor Data Mover.
    if (wave == 0) {
        tdm_load_tile(Wh, (unsigned int)(uintptr_t)&sh_wh[0][0]);
        __builtin_amdgcn_s_wait_tensorcnt(0);
    }
    __syncthreads();

    for (int c = 0; c < NCHUNKS; ++c) {
        int jc = c * CHUNK;
        // Async prefetch of next chunk into the other buffer (overlaps compute).
        if (wave == 0 && c + 1 < NCHUNKS) {
            tdm_load_tile(Wh + (size_t)(jc + CHUNK) * F_OUT,
                          (unsigned int)(uintptr_t)&sh_wh[(c + 1) & 1][0]);
        }
        const float* sh = &sh_wh[c & 1][0];
        const float* dstp = dst + jc;

        #pragma unroll 4
        for (int jj = 0; jj < CHUNK; jj += 4) {
            int j0 = jj + half * 2;            // K-pair base for this lane half
            int2 aa = *(const int2*)(arow + jc + j0);
            float2 dd = *(const float2*)(dstp + j0);   // L2-resident broadcast

            float e0 = lrelu(src_m + dd.x);
            float e1 = lrelu(src_m + dd.y);
            e0 = (aa.x > 0) ? e0 : NEG_INF;    // v_cndmask, no divergence
            e1 = (aa.y > 0) ? e1 : NEG_INF;
            v2f A;
            A.x = __expf(e0 - max_m);
            A.y = __expf(e1 - max_m);

            const float* r0 = sh + j0 * WH_STRIDE;
            const float* r1 = r0 + WH_STRIDE;
            v2f B0, B1, B2, B3;
            B0.x = r0[n];      B0.y = r1[n];
            B1.x = r0[n + 16]; B1.y = r1[n + 16];
            B2.x = r0[n + 32]; B2.y = r1[n + 32];
            B3.x = r0[n + 48]; B3.y = r1[n + 48];

            c0 = __builtin_amdgcn_wmma_f32_16x16x4_f32(false, A, false, B0, (short)0, c0, false, false);
            c1 = __builtin_amdgcn_wmma_f32_16x16x4_f32(false, A, false, B1, (short)0, c1, false, false);
            c2 = __builtin_amdgcn_wmma_f32_16x16x4_f32(false, A, false, B2, (short)0, c2, false, false);
            c3 = __builtin_amdgcn_wmma_f32_16x16x4_f32(false, A, false, B3, (short)0, c3, false, false);
        }

        if (wave == 0) __builtin_amdgcn_s_wait_tensorcnt(0);
        __syncthreads();   // publish next buffer / retire this one
    }

    // Epilogue: divide by row sum, ELU, store.
    #pragma unroll
    for (int r = 0; r < 8; ++r) {
        int row = rowBase + half * 8 + r;
        float inv = 1.0f / rowS[row];
        float v0 = c0[r] * inv;
        float v1 = c1[r] * inv;
        float v2 = c2[r] * inv;
        float v3 = c3[r] * inv;
        float g0 = __expf(v0) - 1.f;
        float g1 = __expf(v1) - 1.f;
        float g2 = __expf(v2) - 1.f;
        float g3 = __expf(v3) - 1.f;
        v0 = (v0 > 0.f) ? v0 : g0;
        v1 = (v1 > 0.f) ? v1 : g1;
        v2 = (v2 > 0.f) ? v2 : g2;
        v3 = (v3 > 0.f) ? v3 : g3;
        float* orow = out + (size_t)row * F_OUT;
        orow[n]      = v0;
        orow[n + 16] = v1;
        orow[n + 32] = v2;
        orow[n + 48] = v3;
    }
}

// ---------------------------------------------------------------------------
extern "C" void kernel_launch(void* const* d_in, const int* in_sizes, int n_in,
                              void* d_out, int out_size, void* d_ws, size_t ws_size,
                              hipStream_t stream) {
    const float* h   = (const float*)d_in[0];
    const int*   adj = (const int*)d_in[1];
    const float* W   = (const float*)d_in[2];
    const float* a   = (const float*)d_in[3];
    float* out = (float*)d_out;

    float* ws   = (float*)d_ws;
    float* Wh   = ws;                         // 8192*64 floats (2 MB)
    float* src  = Wh  + (size_t)N_NODES * F_OUT;
    float* dst  = src + N_NODES;
    float* rowM = dst + N_NODES;
    float* rowS = rowM + N_NODES;

    gat_proj     <<<N_NODES / 4, 256, 0, stream>>>(h, W, a, Wh, src, dst);
    gat_rowstats <<<N_NODES, 256, 0, stream>>>(adj, src, dst, rowM, rowS);
    gat_aggregate<<<N_NODES / 32, K3_THREADS, 0, stream>>>(adj, Wh, src, dst, rowM, rowS, out);
}